// PointNetSetAbstraction_21749714387453
// MI455X (gfx1250) — compile-verified
//
#include <hip/hip_runtime.h>

#define NB 8
#define NP 8192
#define ND 64
#define NS 1024
#define NK 32
#define NM (NB*NS*NK)   // 262144 rows into the MLP
#define RAD2 0.04f
#define BN_EPS 1e-5f

typedef __attribute__((ext_vector_type(2))) float v2f;
typedef __attribute__((ext_vector_type(8))) float v8f;

// ---------------------------------------------------------------------------
// Kernel 1: farthest point sampling. One block per batch, 1024 threads.
// Points live in registers (8 per thread); per-step argmax via LDS tree.
// Also emits new_xyz (workspace, [b][s][3]) and the transposed output (B,3,S).
// ---------------------------------------------------------------------------
__global__ __launch_bounds__(1024)
void fps_kernel(const float* __restrict__ xyz,
                int* __restrict__ fps_idx,
                float* __restrict__ newxyz_ws,
                float* __restrict__ out_newxyz) {
  constexpr int PPT = NP / 1024;
  const int b = blockIdx.x;
  const int tid = threadIdx.x;
  const float* xb = xyz + (size_t)b * 3 * NP;

  float px[PPT], py[PPT], pz[PPT], dist[PPT];
#pragma unroll
  for (int j = 0; j < PPT; ++j) {
    int n = j * 1024 + tid;
    px[j] = xb[n];
    py[j] = xb[NP + n];
    pz[j] = xb[2 * NP + n];
    dist[j] = 1e10f;
  }

  __shared__ float sval[1024];
  __shared__ int   sidx[1024];

  int farthest = 0;
  for (int step = 0; step < NS; ++step) {
    float cx = xb[farthest];
    float cy = xb[NP + farthest];
    float cz = xb[2 * NP + farthest];
    if (tid == 0) {
      fps_idx[b * NS + step] = farthest;
      newxyz_ws[(b * NS + step) * 3 + 0] = cx;
      newxyz_ws[(b * NS + step) * 3 + 1] = cy;
      newxyz_ws[(b * NS + step) * 3 + 2] = cz;
      out_newxyz[(b * 3 + 0) * NS + step] = cx;
      out_newxyz[(b * 3 + 1) * NS + step] = cy;
      out_newxyz[(b * 3 + 2) * NS + step] = cz;
    }
    float best = -1.0f;
    int bestn = 0;
#pragma unroll
    for (int j = 0; j < PPT; ++j) {
      float dx = px[j] - cx, dy = py[j] - cy, dz = pz[j] - cz;
      float d = dx * dx + dy * dy + dz * dz;
      dist[j] = fminf(dist[j], d);
      if (dist[j] > best) { best = dist[j]; bestn = j * 1024 + tid; }
    }
    sval[tid] = best;
    sidx[tid] = bestn;
    __syncthreads();
    for (int off = 512; off > 0; off >>= 1) {
      if (tid < off) {
        float v1 = sval[tid], v2 = sval[tid + off];
        int   i1 = sidx[tid], i2 = sidx[tid + off];
        if (v2 > v1 || (v2 == v1 && i2 < i1)) { sval[tid] = v2; sidx[tid] = i2; }
      }
      __syncthreads();
    }
    farthest = sidx[0];
    __syncthreads();
  }
}

// ---------------------------------------------------------------------------
// Kernel 2: ball query. One thread per (b,s). Ascending index scan collects
// the 32 smallest in-radius indices (== reference's top_k construction),
// padded with the first hit.
// ---------------------------------------------------------------------------
__global__ void ballquery_kernel(const float* __restrict__ xyz,
                                 const float* __restrict__ newxyz_ws,
                                 int* __restrict__ ballidx) {
  int g = blockIdx.x * blockDim.x + threadIdx.x;
  if (g >= NB * NS) return;
  int b = g / NS;
  const float* xb = xyz + (size_t)b * 3 * NP;
  float qx = newxyz_ws[g * 3 + 0];
  float qy = newxyz_ws[g * 3 + 1];
  float qz = newxyz_ws[g * 3 + 2];
  int cnt = 0, first = 0;
  for (int n = 0; n < NP && cnt < NK; ++n) {
    float dx = xb[n] - qx;
    float dy = xb[NP + n] - qy;
    float dz = xb[2 * NP + n] - qz;
    float d = dx * dx + dy * dy + dz * dz;
    if (d <= RAD2) {
      if (cnt == 0) first = n;
      ballidx[(size_t)g * NK + cnt] = n;
      ++cnt;
    }
  }
  for (int k = cnt; k < NK; ++k) ballidx[(size_t)g * NK + k] = first;
}

// ---------------------------------------------------------------------------
// Kernel 3: gather + concat -> featA [M][68] (67 channels + zero pad).
// ---------------------------------------------------------------------------
__global__ void gather_kernel(const float* __restrict__ xyz,
                              const float* __restrict__ points,
                              const float* __restrict__ newxyz_ws,
                              const int* __restrict__ ballidx,
                              float* __restrict__ featA) {
  long i = (long)blockIdx.x * blockDim.x + threadIdx.x;
  if (i >= (long)NM * 68) return;
  int m = (int)(i / 68);
  int c = (int)(i % 68);
  int b = m / (NS * NK);
  int s = (m / NK) % NS;
  int n = ballidx[m];
  float v;
  if (c < 3) {
    v = xyz[(size_t)b * 3 * NP + (size_t)c * NP + n] - newxyz_ws[(b * NS + s) * 3 + c];
  } else if (c < 67) {
    v = points[((size_t)b * ND + (c - 3)) * NP + n];
  } else {
    v = 0.0f;
  }
  featA[(size_t)m * 68 + c] = v;
}

// ---------------------------------------------------------------------------
// Kernel 3b: zero-pad weights W[Cout][Cin] -> Wpad[Cout][CinPad].
// Removes all bounds checks from the GEMM inner loop.
// ---------------------------------------------------------------------------
__global__ void padw_kernel(const float* __restrict__ W, float* __restrict__ Wpad,
                            int Cin, int CinPad, int Cout) {
  int i = blockIdx.x * blockDim.x + threadIdx.x;
  if (i >= Cout * CinPad) return;
  int o = i / CinPad;
  int k = i - o * CinPad;
  Wpad[i] = (k < Cin) ? W[o * Cin + k] : 0.0f;
}

// ---------------------------------------------------------------------------
// Kernel 4: fp32 WMMA GEMM.  H[M][Cout] = A[M][CINPAD] * Wpad[Cout][CINPAD]^T
//           + bias.  One wave per 16x16 tile, V_WMMA_F32_16X16X4_F32 along K.
// CINPAD is compile-time -> K-loop fully unrolled, branch-free inner body.
// A layout: lane holds row M=lane&15, k-pair k0 = kk + 2*(lane>=16).
// B layout: lane holds col N=lane&15, same k-pair (B = Wpad^T).
// ---------------------------------------------------------------------------
template <int CINPAD>
__global__ __launch_bounds__(256)
void gemm_wmma_kernel(const float* __restrict__ A,
                      const float* __restrict__ Wpad,
                      const float* __restrict__ bias,
                      float* __restrict__ H,
                      int Mrows, int Cout) {
  int wavesPerBlock = blockDim.x >> 5;
  int wave = blockIdx.x * wavesPerBlock + (threadIdx.x >> 5);
  int lane = threadIdx.x & 31;
  int ntiles = Cout >> 4;
  int mt = wave / ntiles;
  int nt = wave - mt * ntiles;
  if (mt * 16 >= Mrows) return;

  int row = mt * 16 + (lane & 15);
  int col = nt * 16 + (lane & 15);
  int khalf = (lane >> 4) << 1;   // 0 or 2

  const float* arow = A + (size_t)row * CINPAD + khalf;
  const float* wrow = Wpad + (size_t)col * CINPAD + khalf;

  v8f acc;
  float bv = bias[col];
#pragma unroll
  for (int r = 0; r < 8; ++r) acc[r] = bv;

#pragma unroll
  for (int kk = 0; kk < CINPAD; kk += 4) {
    v2f a  = *(const v2f*)(arow + kk);   // 8B-aligned
    v2f bb = *(const v2f*)(wrow + kk);   // 8B-aligned
    acc = __builtin_amdgcn_wmma_f32_16x16x4_f32(false, a, false, bb,
                                                (short)0, acc, false, false);
  }

  int rbase = mt * 16 + ((lane >> 4) << 3);   // M = r + 8*(lane>=16)
#pragma unroll
  for (int r = 0; r < 8; ++r) {
    H[(size_t)(rbase + r) * Cout + col] = acc[r];
  }
}

// ---------------------------------------------------------------------------
// Kernel 5: per-channel sum / sumsq reduction over H (for batch-norm).
// stats[c] = sum, stats[128+c] = sumsq.
// ---------------------------------------------------------------------------
__global__ void zero_stats_kernel(float* __restrict__ stats) {
  int i = blockIdx.x * blockDim.x + threadIdx.x;
  if (i < 256) stats[i] = 0.0f;
}

__global__ __launch_bounds__(256)
void stats_kernel(const float* __restrict__ H, float* __restrict__ stats,
                  int Mrows, int Cout) {
  __shared__ float ssum[128];
  __shared__ float ssq[128];
  for (int c = threadIdx.x; c < 128; c += blockDim.x) { ssum[c] = 0.0f; ssq[c] = 0.0f; }
  __syncthreads();
  long total = (long)Mrows * Cout;
  for (long i = (long)blockIdx.x * blockDim.x + threadIdx.x; i < total;
       i += (long)gridDim.x * blockDim.x) {
    float v = H[i];
    int c = (int)(i % Cout);
    atomicAdd(&ssum[c], v);
    atomicAdd(&ssq[c], v * v);
  }
  __syncthreads();
  for (int c = threadIdx.x; c < Cout; c += blockDim.x) {
    atomicAdd(&stats[c], ssum[c]);
    atomicAdd(&stats[128 + c], ssq[c]);
  }
}

// ---------------------------------------------------------------------------
// Kernel 6: batch-norm + ReLU, write next layer's A (stride 64).
// ---------------------------------------------------------------------------
__global__ void bn_relu_kernel(const float* __restrict__ H,
                               const float* __restrict__ stats,
                               const float* __restrict__ gamma,
                               const float* __restrict__ beta,
                               float* __restrict__ Out,
                               int Cout, int outStride) {
  long i = (long)blockIdx.x * blockDim.x + threadIdx.x;
  long total = (long)NM * Cout;
  if (i >= total) return;
  int c = (int)(i % Cout);
  long m = i / Cout;
  float invM = 1.0f / (float)NM;
  float mean = stats[c] * invM;
  float var  = stats[128 + c] * invM - mean * mean;
  float inv  = rsqrtf(var + BN_EPS);
  float v = (H[i] - mean) * inv * gamma[c] + beta[c];
  Out[m * outStride + c] = fmaxf(v, 0.0f);
}

// ---------------------------------------------------------------------------
// Kernel 7: final batch-norm + ReLU + maxpool over K, transposed store (B,C,S).
// ---------------------------------------------------------------------------
__global__ void bn_relu_maxpool_kernel(const float* __restrict__ H,
                                       const float* __restrict__ stats,
                                       const float* __restrict__ gamma,
                                       const float* __restrict__ beta,
                                       float* __restrict__ out) {
  int i = blockIdx.x * blockDim.x + threadIdx.x;
  if (i >= NB * NS * 128) return;
  int o = i % 128;
  int bs = i / 128;
  int b = bs / NS;
  int s = bs % NS;
  float invM = 1.0f / (float)NM;
  float mean = stats[o] * invM;
  float var  = stats[128 + o] * invM - mean * mean;
  float inv  = rsqrtf(var + BN_EPS);
  float g = gamma[o], bt = beta[o];
  const float* hrow = H + (size_t)bs * NK * 128 + o;
  float best = 0.0f;   // post-ReLU values are >= 0
#pragma unroll 4
  for (int k = 0; k < NK; ++k) {
    float v = (hrow[(size_t)k * 128] - mean) * inv * g + bt;
    best = fmaxf(best, fmaxf(v, 0.0f));
  }
  out[(size_t)(b * 128 + o) * NS + s] = best;
}

// ---------------------------------------------------------------------------
extern "C" void kernel_launch(void* const* d_in, const int* in_sizes, int n_in,
                              void* d_out, int out_size, void* d_ws, size_t ws_size,
                              hipStream_t stream) {
  const float* xyz    = (const float*)d_in[0];
  const float* points = (const float*)d_in[1];
  const float* w0 = (const float*)d_in[2];
  const float* b0 = (const float*)d_in[3];
  const float* g0 = (const float*)d_in[4];
  const float* t0 = (const float*)d_in[5];
  const float* w1 = (const float*)d_in[6];
  const float* b1 = (const float*)d_in[7];
  const float* g1 = (const float*)d_in[8];
  const float* t1 = (const float*)d_in[9];
  const float* w2 = (const float*)d_in[10];
  const float* b2 = (const float*)d_in[11];
  const float* g2 = (const float*)d_in[12];
  const float* t2 = (const float*)d_in[13];
  float* out = (float*)d_out;

  char* ws = (char*)d_ws;
  int*   fpsidx  = (int*)(ws);                               //  32 KB
  float* newxyz  = (float*)(ws + 32768);                     //  96 KB
  int*   ballidx = (int*)(ws + 131072);                      //   1 MB
  float* featA   = (float*)(ws + 1179648);                   //  71 MB  [M][68]
  size_t offHB   = 1179648 + (size_t)NM * 68 * 4;
  float* hB      = (float*)(ws + offHB);                     // 134 MB  [M][128]
  size_t offSt   = offHB + (size_t)NM * 128 * 4;
  float* stats   = (float*)(ws + offSt);                     //   1 KB
  float* wpad    = (float*)(ws + offSt + 1024);              //  35 KB  [Cout][CinPad]

  // 1) FPS (also writes new_xyz output, transposed)
  fps_kernel<<<NB, 1024, 0, stream>>>(xyz, fpsidx, newxyz, out);

  // 2) Ball query
  ballquery_kernel<<<(NB * NS + 255) / 256, 256, 0, stream>>>(xyz, newxyz, ballidx);

  // 3) Gather + concat
  {
    long total = (long)NM * 68;
    gather_kernel<<<(int)((total + 255) / 256), 256, 0, stream>>>(xyz, points, newxyz,
                                                                  ballidx, featA);
  }

  auto launch_stats = [&](const float* H, int Cout) {
    zero_stats_kernel<<<1, 256, 0, stream>>>(stats);
    stats_kernel<<<1024, 256, 0, stream>>>(H, stats, NM, Cout);
  };

  // Layer 0: 67 -> 64  (CinPad 68)
  padw_kernel<<<(64 * 68 + 255) / 256, 256, 0, stream>>>(w0, wpad, 67, 68, 64);
  {
    int waves = (NM / 16) * (64 / 16);
    gemm_wmma_kernel<68><<<waves / 8, 256, 0, stream>>>(featA, wpad, b0, hB, NM, 64);
  }
  launch_stats(hB, 64);
  bn_relu_kernel<<<(int)(((long)NM * 64 + 255) / 256), 256, 0, stream>>>(
      hB, stats, g0, t0, featA, 64, 64);

  // Layer 1: 64 -> 64  (CinPad 64, no pad needed but keep uniform path)
  padw_kernel<<<(64 * 64 + 255) / 256, 256, 0, stream>>>(w1, wpad, 64, 64, 64);
  {
    int waves = (NM / 16) * (64 / 16);
    gemm_wmma_kernel<64><<<waves / 8, 256, 0, stream>>>(featA, wpad, b1, hB, NM, 64);
  }
  launch_stats(hB, 64);
  bn_relu_kernel<<<(int)(((long)NM * 64 + 255) / 256), 256, 0, stream>>>(
      hB, stats, g1, t1, featA, 64, 64);

  // Layer 2: 64 -> 128
  padw_kernel<<<(128 * 64 + 255) / 256, 256, 0, stream>>>(w2, wpad, 64, 64, 128);
  {
    int waves = (NM / 16) * (128 / 16);
    gemm_wmma_kernel<64><<<waves / 8, 256, 0, stream>>>(featA, wpad, b2, hB, NM, 128);
  }
  launch_stats(hB, 128);

  // Final: BN + ReLU + maxpool over K, transposed output after new_xyz block
  bn_relu_maxpool_kernel<<<(NB * NS * 128 + 255) / 256, 256, 0, stream>>>(
      hB, stats, g2, t2, out + NB * 3 * NS);
}